// RTNTM_52819507806361
// MI455X (gfx1250) — compile-verified
//
#include <hip/hip_runtime.h>
#include <math.h>

// Problem constants (match reference)
#define B_   32
#define T_   128
#define V_   97
#define D_   512
#define N_   128
#define M_   64
#define H_   8
#define DH_  64
#define W_   8
#define K_   3
#define RPL_ 70    // M + 3 + K
#define WPL_ 198   // RPL + 2*M
#define DM_  576   // D + M
#define FF_  2048  // 4*D

typedef __attribute__((ext_vector_type(2))) float v2f;
typedef __attribute__((ext_vector_type(8))) float v8f;

__device__ __forceinline__ float softplusf(float x) {
  // stable log1p(exp(x)) = max(x,0) + log1p(exp(-|x|))
  return fmaxf(x, 0.f) + log1pf(expf(-fabsf(x)));
}
__device__ __forceinline__ float sigmoidf(float x) {
  return 1.f / (1.f + expf(-x));
}
__device__ __forceinline__ float gelu_exact(float x) {
  return 0.5f * x * (1.f + erff(x * 0.70710678118654752f));
}

// ---------------------------------------------------------------------------
// Phase 1: tok = emb[x_t]; r = rw @ memory; wide = [tok | r]
// grid = B_, block = 256
// ---------------------------------------------------------------------------
__global__ void k_read_wide(const int* __restrict__ x,
                            const float* __restrict__ emb,
                            const float* __restrict__ rw,
                            const float* __restrict__ mem,
                            float* __restrict__ wide, int t) {
  int b = blockIdx.x, tid = threadIdx.x;
  int tok = x[b * T_ + t];
  for (int d = tid; d < D_; d += 256)
    wide[b * DM_ + d] = emb[tok * D_ + d];
  if (tid < M_) {
    float acc = 0.f;
    const float* mrow = mem + (size_t)b * N_ * M_ + tid;
    for (int n = 0; n < N_; ++n)
      acc += rw[b * N_ + n] * mrow[(size_t)n * M_];
    wide[b * DM_ + D_ + tid] = acc;
  }
}

// ---------------------------------------------------------------------------
// Generic f32-WMMA GEMM:  Out[32, Ncols] = X[32, Kdim] * W^T + bias
// W is row-major [Ncols, Kdim] (as in all reference weight matrices).
//
// Each wave owns one 16-row tile x TWO adjacent 16-col tiles, so a single
// A fragment feeds two v_wmma ops (A reuse, 3 b64 loads per 2 WMMAs).
// Out-of-range columns are CLAMPED for loads (branch-free inner loop: no
// EXEC save/restore) -- garbage stays confined to columns we never store,
// since C[:,n] depends only on the B elements of the lanes owning column n.
//
// V_WMMA_F32_16X16X4_F32 VGPR layouts (ISA 7.12.2):
//   A (16x4):  lane l -> A[l&15, 2*(l>>4) + {0,1}]   (.x, .y)
//   B (4x16):  lane l -> B[2*(l>>4) + {0,1}, l&15]   (.x, .y)  [B[k,n] = W[n,k]]
//   C (16x16): elem v -> row v + 8*(l>>4), col l&15
// act: 0 = none, 1 = exact GELU
// blockDim = 256 (8 waves); wave-uniform early-out for tail waves.
// ---------------------------------------------------------------------------
__global__ void k_gemm(const float* __restrict__ X, int ldx,
                       const float* __restrict__ Wm,
                       const float* __restrict__ bias,
                       float* __restrict__ Out, int ldo,
                       int Kdim, int Ncols, int act,
                       int pairCols, int totalPairs) {
  int pair = blockIdx.x * (blockDim.x >> 5) + (threadIdx.x >> 5);
  if (pair >= totalPairs) return;           // wave-uniform
  int tr = pair / pairCols;                 // 0..1 (rows 0-15 / 16-31)
  int pc = pair % pairCols;
  int l = threadIdx.x & 31;                 // lane in wave (EXEC all ones)
  int half = l >> 4;                        // 0/1
  int lan = l & 15;

  int arow = tr * 16 + lan;                 // A row this lane supplies
  int n0 = pc * 32 + lan;                   // column tile 0
  int n1 = n0 + 16;                         // column tile 1
  bool v0 = n0 < Ncols, v1 = n1 < Ncols;
  int n0c = v0 ? n0 : (Ncols - 1);          // clamp for branch-free loads
  int n1c = v1 ? n1 : (Ncols - 1);

  const float* xr = X + (size_t)arow * ldx + 2 * half;
  const float* w0 = Wm + (size_t)n0c * Kdim + 2 * half;
  const float* w1 = Wm + (size_t)n1c * Kdim + 2 * half;

  float b0 = bias[n0c], b1 = bias[n1c];
  v8f c0, c1;
#pragma unroll
  for (int v = 0; v < 8; ++v) { c0[v] = b0; c1[v] = b1; }

  for (int k = 0; k < Kdim; k += 4) {
    v2f a, f0, f1;
    a.x = xr[k];   a.y = xr[k + 1];
    f0.x = w0[k];  f0.y = w0[k + 1];
    f1.x = w1[k];  f1.y = w1[k + 1];
    c0 = __builtin_amdgcn_wmma_f32_16x16x4_f32(false, a, false, f0,
                                               (short)0, c0, false, false);
    c1 = __builtin_amdgcn_wmma_f32_16x16x4_f32(false, a, false, f1,
                                               (short)0, c1, false, false);
  }

#pragma unroll
  for (int v = 0; v < 8; ++v) {
    int row = tr * 16 + v + 8 * half;
    float x0 = c0[v], x1 = c1[v];
    if (act == 1) { x0 = gelu_exact(x0); x1 = gelu_exact(x1); }
    if (v0) Out[(size_t)row * ldo + n0] = x0;
    if (v1) Out[(size_t)row * ldo + n1] = x1;
  }
}

// ---------------------------------------------------------------------------
// Windowed attention over the circular K/V caches.
// grid = 1, block = 256: one thread per (b, h).
// Slot s of the cache is valid iff s < min(t+1, W).
// ---------------------------------------------------------------------------
__global__ void k_attn(const float* __restrict__ q,
                       const float* __restrict__ kc,
                       const float* __restrict__ vc,
                       float* __restrict__ ctx, int t) {
  int id = threadIdx.x;              // 0..255
  int b = id >> 3, h = id & 7;
  const float scale = 0.125f;        // 1/sqrt(DH)
  int nvalid = (t + 1 < W_) ? (t + 1) : W_;
  const float* qv = q + (size_t)b * D_ + h * DH_;

  float s[W_];
  float mx = -3.4e38f;
  for (int w = 0; w < W_; ++w) {
    if (w < nvalid) {
      const float* kv = kc + ((size_t)w * B_ + b) * D_ + h * DH_;
      float dot = 0.f;
      for (int e = 0; e < DH_; ++e) dot += qv[e] * kv[e];
      s[w] = dot * scale;
      mx = fmaxf(mx, s[w]);
    }
  }
  float denom = 0.f;
  for (int w = 0; w < nvalid; ++w) { s[w] = expf(s[w] - mx); denom += s[w]; }
  float inv = 1.f / denom;

  for (int e = 0; e < DH_; ++e) {
    float acc = 0.f;
    for (int w = 0; w < nvalid; ++w)
      acc += s[w] * vc[((size_t)w * B_ + b) * D_ + h * DH_ + e];
    ctx[(size_t)b * D_ + h * DH_ + e] = acc * inv;
  }
}

// ---------------------------------------------------------------------------
// Fused residual add + LayerNorm: out = LN(a + c) * g + beta
// grid = B_, block = 256 (2 elements / thread)
// ---------------------------------------------------------------------------
__global__ void k_ln(const float* __restrict__ a, const float* __restrict__ c,
                     const float* __restrict__ g, const float* __restrict__ be,
                     float* __restrict__ out) {
  int b = blockIdx.x, tid = threadIdx.x;
  __shared__ float s1[256], s2[256];
  float v0 = a[b * D_ + tid] + c[b * D_ + tid];
  float v1 = a[b * D_ + tid + 256] + c[b * D_ + tid + 256];
  s1[tid] = v0 + v1;
  s2[tid] = v0 * v0 + v1 * v1;
  __syncthreads();
  for (int st = 128; st > 0; st >>= 1) {
    if (tid < st) { s1[tid] += s1[tid + st]; s2[tid] += s2[tid + st]; }
    __syncthreads();
  }
  float mean = s1[0] * (1.f / D_);
  float var = s2[0] * (1.f / D_) - mean * mean;
  float inv = rsqrtf(var + 1e-5f);
  out[b * D_ + tid]       = (v0 - mean) * inv * g[tid]       + be[tid];
  out[b * D_ + tid + 256] = (v1 - mean) * inv * g[tid + 256] + be[tid + 256];
}

// ---------------------------------------------------------------------------
// NTM addressing: p row = [key(64) | beta | gate | shift(3) | gamma | ...]
// grid = B_, block = N_ (128). Reads prev w from wvec, overwrites with new w.
// ---------------------------------------------------------------------------
__global__ void k_addr(const float* __restrict__ p, int ldp,
                       const float* __restrict__ mem,
                       float* __restrict__ wvec) {
  const float eps = 1e-12f;
  int b = blockIdx.x, n = threadIdx.x;
  __shared__ float key[M_];
  __shared__ float red[N_];
  __shared__ float wg[N_];

  if (n < M_) key[n] = p[b * ldp + n];
  float prev = wvec[b * N_ + n];
  __syncthreads();

  // ||key||
  red[n] = (n < M_) ? key[n] * key[n] : 0.f;
  __syncthreads();
  for (int st = 64; st > 0; st >>= 1) {
    if (n < st) red[n] += red[n + st];
    __syncthreads();
  }
  float knorm = sqrtf(red[0]) + eps;
  __syncthreads();

  // cosine similarity of memory row n with key
  float dot = 0.f, nr = 0.f;
  const float* mrow = mem + ((size_t)b * N_ + n) * M_;
  for (int m = 0; m < M_; ++m) {
    float mv = mrow[m];
    dot += mv * key[m];
    nr += mv * mv;
  }
  float cosv = dot / ((sqrtf(nr) + eps) * knorm);
  float beta = softplusf(p[b * ldp + M_]);
  float aa = beta * cosv;

  // softmax over N
  red[n] = aa;
  __syncthreads();
  for (int st = 64; st > 0; st >>= 1) {
    if (n < st) red[n] = fmaxf(red[n], red[n + st]);
    __syncthreads();
  }
  float mx = red[0];
  __syncthreads();
  float ev = expf(aa - mx);
  red[n] = ev;
  __syncthreads();
  for (int st = 64; st > 0; st >>= 1) {
    if (n < st) red[n] += red[n + st];
    __syncthreads();
  }
  float wc = ev / red[0];
  __syncthreads();

  // interpolate with previous weights
  float g = sigmoidf(p[b * ldp + M_ + 1]);
  float wgv = g * wc + (1.f - g) * prev;
  wg[n] = wgv;

  // shift softmax (3 taps, per-thread)
  float s0 = p[b * ldp + M_ + 2];
  float s1 = p[b * ldp + M_ + 3];
  float s2 = p[b * ldp + M_ + 4];
  float sm = fmaxf(s0, fmaxf(s1, s2));
  float e0 = expf(s0 - sm), e1 = expf(s1 - sm), e2 = expf(s2 - sm);
  float esum = e0 + e1 + e2;
  __syncthreads();
  // shifted[i] = s[0]*wg[(i+1)%N] + s[1]*wg[i] + s[2]*wg[(i-1)%N]
  float shifted = (e0 * wg[(n + 1) & (N_ - 1)] + e1 * wg[n] +
                   e2 * wg[(n + N_ - 1) & (N_ - 1)]) / esum;

  // sharpen
  float gp = 1.f + softplusf(p[b * ldp + M_ + 2 + K_]);
  float wt = powf(shifted + eps, gp);
  red[n] = wt;
  __syncthreads();
  for (int st = 64; st > 0; st >>= 1) {
    if (n < st) red[n] += red[n + st];
    __syncthreads();
  }
  wvec[b * N_ + n] = wt / (red[0] + eps);
}

// ---------------------------------------------------------------------------
// Memory update: mem = mem*(1 - ww*erase) + ww*add
// grid = B_*N_, block = M_
// ---------------------------------------------------------------------------
__global__ void k_memupd(float* __restrict__ mem,
                         const float* __restrict__ ww,
                         const float* __restrict__ wp, int ldp) {
  int blk = blockIdx.x;
  int b = blk >> 7, n = blk & (N_ - 1);
  int m = threadIdx.x;
  float w = ww[b * N_ + n];
  float er = sigmoidf(wp[b * ldp + RPL_ + m]);
  float ad = tanhf(wp[b * ldp + RPL_ + M_ + m]);
  size_t idx = ((size_t)b * N_ + n) * M_ + m;
  float mv = mem[idx];
  mem[idx] = mv * (1.f - w * er) + w * ad;
}

// Initialize rw/ww to one-hot at position 0.
__global__ void k_init_w(float* __restrict__ rw, float* __restrict__ ww) {
  int i = blockIdx.x * blockDim.x + threadIdx.x;
  if (i < B_ * N_) {
    float v = ((i & (N_ - 1)) == 0) ? 1.f : 0.f;
    rw[i] = v;
    ww[i] = v;
  }
}

// ---------------------------------------------------------------------------
static void gemm(hipStream_t s, const float* X, int ldx, const float* Wm,
                 const float* bias, float* Out, int ldo, int Kdim, int Ncols,
                 int act) {
  int ct = (Ncols + 15) / 16;
  int pairCols = (ct + 1) / 2;
  int totalPairs = 2 * pairCols;            // 2 row tiles x pairCols
  int blocks = (totalPairs + 7) / 8;        // 8 waves / block
  k_gemm<<<dim3(blocks), dim3(256), 0, s>>>(X, ldx, Wm, bias, Out, ldo, Kdim,
                                            Ncols, act, pairCols, totalPairs);
}

extern "C" void kernel_launch(void* const* d_in, const int* in_sizes, int n_in,
                              void* d_out, int out_size, void* d_ws,
                              size_t ws_size, hipStream_t stream) {
  (void)in_sizes; (void)n_in; (void)out_size; (void)ws_size;
  const int*   x       = (const int*)d_in[0];
  const float* emb     = (const float*)d_in[1];
  const float* W_in    = (const float*)d_in[2];
  const float* b_in    = (const float*)d_in[3];
  const float* W_qkv   = (const float*)d_in[4];
  const float* b_qkv   = (const float*)d_in[5];
  const float* W_o     = (const float*)d_in[6];
  const float* b_o     = (const float*)d_in[7];
  const float* ln1_g   = (const float*)d_in[8];
  const float* ln1_b   = (const float*)d_in[9];
  const float* W_ff1   = (const float*)d_in[10];
  const float* b_ff1   = (const float*)d_in[11];
  const float* W_ff2   = (const float*)d_in[12];
  const float* b_ff2   = (const float*)d_in[13];
  const float* ln2_g   = (const float*)d_in[14];
  const float* ln2_b   = (const float*)d_in[15];
  const float* W_tok   = (const float*)d_in[16];
  const float* b_tok   = (const float*)d_in[17];
  const float* W_read  = (const float*)d_in[18];
  const float* b_read  = (const float*)d_in[19];
  const float* W_write = (const float*)d_in[20];
  const float* b_write = (const float*)d_in[21];
  float* dout = (float*)d_out;

  const float* Wq = W_qkv;
  const float* Wk = W_qkv + (size_t)D_ * D_;
  const float* Wv = W_qkv + (size_t)2 * D_ * D_;
  const float* bq = b_qkv;
  const float* bk = b_qkv + D_;
  const float* bv = b_qkv + 2 * D_;

  // Workspace layout (floats)
  float* wsf = (float*)d_ws;
  size_t o = 0;
  float* memory = wsf + o; o += (size_t)B_ * N_ * M_;   // 262144
  float* rw     = wsf + o; o += (size_t)B_ * N_;
  float* ww     = wsf + o; o += (size_t)B_ * N_;
  float* kc     = wsf + o; o += (size_t)W_ * B_ * D_;   // K cache (circular)
  float* vc     = wsf + o; o += (size_t)W_ * B_ * D_;   // V cache (circular)
  float* wide   = wsf + o; o += (size_t)B_ * DM_;
  float* xp     = wsf + o; o += (size_t)B_ * D_;
  float* qb     = wsf + o; o += (size_t)B_ * D_;
  float* ctxb   = wsf + o; o += (size_t)B_ * D_;
  float* atno   = wsf + o; o += (size_t)B_ * D_;
  float* x2     = wsf + o; o += (size_t)B_ * D_;
  float* ffa    = wsf + o; o += (size_t)B_ * FF_;
  float* ffo    = wsf + o; o += (size_t)B_ * D_;
  float* outb   = wsf + o; o += (size_t)B_ * D_;
  float* rp     = wsf + o; o += (size_t)B_ * RPL_;
  float* wp     = wsf + o; o += (size_t)B_ * WPL_;

  // Reset recurrent state (must be identical every call)
  hipMemsetAsync(memory, 0, (size_t)B_ * N_ * M_ * sizeof(float), stream);
  k_init_w<<<dim3((B_ * N_ + 255) / 256), dim3(256), 0, stream>>>(rw, ww);

  for (int t = 0; t < T_; ++t) {
    int slot = t & (W_ - 1);
    k_read_wide<<<dim3(B_), dim3(256), 0, stream>>>(x, emb, rw, memory, wide, t);
    gemm(stream, wide, DM_, W_in, b_in, xp, D_, DM_, D_, 0);
    gemm(stream, xp, D_, Wq, bq, qb, D_, D_, D_, 0);
    gemm(stream, xp, D_, Wk, bk, kc + (size_t)slot * B_ * D_, D_, D_, D_, 0);
    gemm(stream, xp, D_, Wv, bv, vc + (size_t)slot * B_ * D_, D_, D_, D_, 0);
    k_attn<<<dim3(1), dim3(256), 0, stream>>>(qb, kc, vc, ctxb, t);
    gemm(stream, ctxb, D_, W_o, b_o, atno, D_, D_, D_, 0);
    k_ln<<<dim3(B_), dim3(256), 0, stream>>>(xp, atno, ln1_g, ln1_b, x2);
    gemm(stream, x2, D_, W_ff1, b_ff1, ffa, FF_, D_, FF_, 1);   // fused GELU
    gemm(stream, ffa, FF_, W_ff2, b_ff2, ffo, D_, FF_, D_, 0);
    k_ln<<<dim3(B_), dim3(256), 0, stream>>>(x2, ffo, ln2_g, ln2_b, outb);
    gemm(stream, outb, D_, W_tok, b_tok, dout + (size_t)t * V_, T_ * V_, D_, V_, 0);
    gemm(stream, outb, D_, W_read, b_read, rp, RPL_, D_, RPL_, 0);
    gemm(stream, outb, D_, W_write, b_write, wp, WPL_, D_, WPL_, 0);
    k_addr<<<dim3(B_), dim3(N_), 0, stream>>>(rp, RPL_, memory, rw);
    k_addr<<<dim3(B_), dim3(N_), 0, stream>>>(wp, WPL_, memory, ww);
    k_memupd<<<dim3(B_ * N_), dim3(M_), 0, stream>>>(memory, ww, wp, WPL_);
  }
}